// CenterLoss_61314953118306
// MI455X (gfx1250) — compile-verified
//
#include <hip/hip_runtime.h>
#include <hip/hip_bf16.h>
#include <stdint.h>

// ---------------------------------------------------------------------------
// CenterLoss on MI455X (gfx1250, wave32).
//   loss = ( sum_n clip(||f_n - c_{argmax_n}||^2, 1e-12, 1e12)
//            + N*(C-1)*1e-12 ) / N
// Kernel 1: per-row argmax over predicts (108.5 MB -> the bandwidth roofline)
// Kernel 2: per-16-row tile, diagonal of f*c^T via v_wmma_f32_16x16x4_f32
// Kernel 3: deterministic single-block tree reduction
// ---------------------------------------------------------------------------

#define FEAT_DIM 96

typedef __attribute__((ext_vector_type(2))) float v2f;
typedef __attribute__((ext_vector_type(8))) float v8f;

// -------------------------- Kernel 1: argmax -------------------------------
// One wave32 per row of predicts[nrows][ncls]. 128-bit coalesced body with
// scalar head/tail for 16B alignment (row byte offset is 4*row mod 16).
// First-index tie-break to match jnp.argmax.
__global__ __launch_bounds__(256) void cl_argmax_rows(
    const float* __restrict__ pred, int* __restrict__ labels,
    int nrows, int ncls) {
  const int wave = (blockIdx.x * blockDim.x + threadIdx.x) >> 5;
  const int lane = threadIdx.x & 31;
  if (wave >= nrows) return;

  const float* __restrict__ rowp = pred + (size_t)wave * (size_t)ncls;

  float bv = -__builtin_inff();
  int   bi = 0;

  // Head: bring pointer to 16B alignment.
  const int mis = (int)(((uintptr_t)rowp >> 2) & 3);  // misalignment in floats
  int h = (4 - mis) & 3;
  if (h > ncls) h = ncls;
  if (lane < h) {
    float v = rowp[lane];
    if (v > bv) { bv = v; bi = lane; }
  }

  // Body: float4 per lane, 512B contiguous per wave-iteration.
  const int n4 = (ncls - h) >> 2;
  const float4* __restrict__ q = (const float4*)(rowp + h);
#pragma unroll 4
  for (int j = lane; j < n4; j += 32) {
    float4 v = q[j];
    int base = h + 4 * j;
    if (v.x > bv) { bv = v.x; bi = base; }
    if (v.y > bv) { bv = v.y; bi = base + 1; }
    if (v.z > bv) { bv = v.z; bi = base + 2; }
    if (v.w > bv) { bv = v.w; bi = base + 3; }
  }

  // Tail.
  for (int idx = h + 4 * n4 + lane; idx < ncls; idx += 32) {
    float v = rowp[idx];
    if (v > bv) { bv = v; bi = idx; }
  }

  // Wave32 reduction; on value ties keep the smaller index (first occurrence).
  for (int off = 16; off > 0; off >>= 1) {
    float ov = __shfl_xor(bv, off, 32);
    int   oi = __shfl_xor(bi, off, 32);
    if (ov > bv || (ov == bv && oi < bi)) { bv = ov; bi = oi; }
  }
  if (lane == 0) labels[wave] = bi;
}

// ------------------- Kernel 2: WMMA diagonal distances ---------------------
// One wave per 16 rows. A = feats tile (16 x 4 per step), B = gathered
// centers (4 x 16 per step); 24 chained v_wmma_f32_16x16x4_f32 give the
// 16x16 f.c^T tile; we keep only the diagonal (f_n . c_{label_n}).
//
// f32 16x4 A layout (ISA 7.12.2): lanes 0-15 -> M=lane, VGPR0/1 = K0/K1;
// lanes 16-31 -> M=lane-16, VGPR0/1 = K2/K3. B (4x16) mirrors it with rows
// striped across lanes. C/D: VGPR v, lanes 0-15 -> (M=v, N=lane),
// lanes 16-31 -> (M=v+8, N=lane-16).
__global__ __launch_bounds__(256) void cl_tile_dist(
    const float* __restrict__ feats, const float* __restrict__ centers,
    const int* __restrict__ labels, float* __restrict__ rowdist,
    int ntiles) {
  const int wave = (blockIdx.x * blockDim.x + threadIdx.x) >> 5;
  const int lane = threadIdx.x & 31;
  if (wave >= ntiles) return;  // wave-uniform; surviving waves have full EXEC

  const int n0 = wave * 16;
  const int r  = lane & 15;          // row within tile (both A's M and B's N)
  const int kh = (lane >> 4) << 1;   // K sub-offset: 0 for lanes 0-15, 2 else
  const int row = n0 + r;
  const int lbl = labels[row];

  const float* __restrict__ fp = feats   + (size_t)row * FEAT_DIM;
  const float* __restrict__ cp = centers + (size_t)lbl * FEAT_DIM;

  v8f acc = {};
#pragma unroll
  for (int k0 = 0; k0 < FEAT_DIM; k0 += 4) {
    v2f a, b;
    a.x = fp[k0 + kh];     a.y = fp[k0 + kh + 1];
    b.x = cp[k0 + kh];     b.y = cp[k0 + kh + 1];
    acc = __builtin_amdgcn_wmma_f32_16x16x4_f32(
        /*neg_a=*/false, a, /*neg_b=*/false, b,
        /*c_mod=*/(short)0, acc, /*reuse_a=*/false, /*reuse_b=*/false);
  }

  // Row norms (every lane computes its own row's ||f||^2 and ||c||^2).
  float sf = 0.f, sc = 0.f;
#pragma unroll 8
  for (int k = 0; k < FEAT_DIM; ++k) {
    float fv = fp[k], cv = cp[k];
    sf = fmaf(fv, fv, sf);
    sc = fmaf(cv, cv, sc);
  }

  // Diagonal M==N lives in lanes 0-7 (comp=lane) and 24-31 (comp=lane-24).
  const bool isDiag = (lane < 8) | (lane >= 24);
  const int  comp   = (lane < 8) ? lane : (lane - 24);
  float dot = 0.f;
#pragma unroll
  for (int i = 0; i < 8; ++i) dot = (comp == i) ? acc[i] : dot;

  if (isDiag) {
    float d = sf + sc - 2.0f * dot;
    d = fminf(fmaxf(d, 1e-12f), 1e12f);
    rowdist[row] = d;
  }
}

// ---------------- Kernel 3: deterministic final reduction ------------------
__global__ __launch_bounds__(256) void cl_reduce(
    const float* __restrict__ rowdist, float* __restrict__ out,
    int nrows, float extra) {
  __shared__ float s[256];
  const int tid = threadIdx.x;
  float a = 0.f;
  for (int i = tid; i < nrows; i += 256) a += rowdist[i];
  s[tid] = a;
  __syncthreads();
  for (int off = 128; off > 0; off >>= 1) {
    if (tid < off) s[tid] += s[tid + off];
    __syncthreads();
  }
  if (tid == 0) out[0] = (s[0] + extra) / (float)nrows;
}

// ---------------------------------------------------------------------------
extern "C" void kernel_launch(void* const* d_in, const int* in_sizes, int n_in,
                              void* d_out, int out_size, void* d_ws, size_t ws_size,
                              hipStream_t stream) {
  const float* feats   = (const float*)d_in[0];  // [B*T, 96]
  const float* pred    = (const float*)d_in[1];  // [B*T, C]
  const float* centers = (const float*)d_in[2];  // [C, 96]
  float* out = (float*)d_out;

  const int nrows = in_sizes[0] / FEAT_DIM;            // 4096
  const int ncls  = in_sizes[2] / FEAT_DIM;            // 6625

  // Workspace: labels[nrows] (int) then rowdist[nrows] (float), 256B aligned.
  int*   labels  = (int*)d_ws;
  size_t off     = (((size_t)nrows * sizeof(int)) + 255u) & ~(size_t)255u;
  float* rowdist = (float*)((char*)d_ws + off);

  // 1) argmax: one wave per row, 8 waves per block.
  {
    int waves_per_block = 256 / 32;
    int blocks = (nrows + waves_per_block - 1) / waves_per_block;
    cl_argmax_rows<<<blocks, 256, 0, stream>>>(pred, labels, nrows, ncls);
  }

  // 2) WMMA tile distances: one wave per 16 rows.
  {
    int ntiles = nrows / 16;                    // 256
    int waves_per_block = 256 / 32;
    int blocks = (ntiles + waves_per_block - 1) / waves_per_block;
    cl_tile_dist<<<blocks, 256, 0, stream>>>(feats, centers, labels, rowdist, ntiles);
  }

  // 3) Deterministic reduction; add N*(C-1)*1e-12 from clip-after-mask.
  {
    float extra = (float)((double)nrows * (double)(ncls - 1) * 1e-12);
    cl_reduce<<<1, 256, 0, stream>>>(rowdist, out, nrows, extra);
  }
}